// ScaledDotProductSparseAttention_66666482369204
// MI455X (gfx1250) — compile-verified
//
#include <hip/hip_runtime.h>
#include <hip/hip_bf16.h>

typedef float v2f __attribute__((ext_vector_type(2)));
typedef float v4f __attribute__((ext_vector_type(4)));
typedef float v8f __attribute__((ext_vector_type(8)));

#define BATCH 64
#define LQ 1024
#define LK 1024
#define DH 64
#define QTILE 16
#define NTHREADS 256
#define NWAVES 8

// LDS layout (floats): z[16*1024] | tau[16] | partials[8*256]
#define Z_ELEMS (QTILE * LK)
#define TAU_OFF Z_ELEMS
#define P_OFF (Z_ELEMS + 16)
#define SMEM_FLOATS (Z_ELEMS + 16 + NWAVES * 256)
#define SMEM_BYTES (SMEM_FLOATS * 4)

__device__ __forceinline__ float wred_sum_f(float v) {
#pragma unroll
  for (int o = 16; o > 0; o >>= 1) v += __shfl_xor(v, o, 32);
  return v;
}
__device__ __forceinline__ int wred_sum_i(int v) {
#pragma unroll
  for (int o = 16; o > 0; o >>= 1) v += __shfl_xor(v, o, 32);
  return v;
}

__global__ void sparsemax_attn_kernel(const float* __restrict__ q,
                                      const float* __restrict__ k,
                                      const float* __restrict__ v,
                                      const unsigned char* __restrict__ mask,
                                      float* __restrict__ out,
                                      float* __restrict__ attn) {
  extern __shared__ float smem[];
  float* zsm   = smem;
  float* tausm = smem + TAU_OFF;
  float* psm   = smem + P_OFF;
  v4f*   zsm4  = reinterpret_cast<v4f*>(smem);

  const int b  = blockIdx.y;
  const int q0 = blockIdx.x * QTILE;
  const int tid  = threadIdx.x;
  const int lane = tid & 31;
  const int wv   = tid >> 5;     // 0..7
  const int ln   = lane & 15;    // N / M index within fragment
  const int kh   = lane >> 4;    // K-half selector (f32 16x16x4 layout)

  // ---------------- Phase 1: scores = (Q/8) @ K^T via WMMA f32 16x16x4 ----
  // A fragment (Q rows q0..q0+15), 16 k-steps covering D=64.
  // Layout: lane l holds A[m=l&15][ks*4 + 2*(l>>4) + {0,1}] in {a.x, a.y}
  v2f a[16];
  {
    const float* qrow = q + ((size_t)b * LQ + q0 + ln) * DH;
#pragma unroll
    for (int s = 0; s < 16; ++s) {
      a[s].x = qrow[s * 4 + 2 * kh]     * 0.125f;   // 1/TEMPERATURE
      a[s].y = qrow[s * 4 + 2 * kh + 1] * 0.125f;
    }
  }
  // Each wave computes 8 contiguous 16-col key tiles (128 cols per wave).
  for (int t = 0; t < 8; ++t) {
    const int n0 = (wv * 8 + t) * QTILE;
    const float* krow = k + ((size_t)b * LK + n0 + ln) * DH;
    v8f c = {0.f, 0.f, 0.f, 0.f, 0.f, 0.f, 0.f, 0.f};
#pragma unroll
    for (int s = 0; s < 16; ++s) {
      v2f bb;
      bb.x = krow[s * 4 + 2 * kh];
      bb.y = krow[s * 4 + 2 * kh + 1];
      c = __builtin_amdgcn_wmma_f32_16x16x4_f32(false, a[s], false, bb,
                                                (short)0, c, false, false);
    }
    // D layout: c[j] is element (M = j + 8*kh, N = n0 + ln)
#pragma unroll
    for (int j = 0; j < 8; ++j) zsm[(j + 8 * kh) * LK + n0 + ln] = c[j];
  }
  __syncthreads();

  // ------- Phase 2: apply mask (True -> -3e38), 4 bools per uint load -----
  // mask is streaming/read-once: non-temporal loads keep K/V hot in L2.
  {
    const unsigned int* mrow =
        reinterpret_cast<const unsigned int*>(mask + ((size_t)b * LQ + q0) * LK);
#pragma unroll 4
    for (int i = 0; i < Z_ELEMS / (4 * NTHREADS); ++i) {
      const int e4 = tid + NTHREADS * i;             // float4 index
      const unsigned int m4 = __builtin_nontemporal_load(&mrow[e4]);
      if (m4) {
        v4f z4 = zsm4[e4];
        if (m4 & 0x000000FFu) z4.x = -3.0e38f;
        if (m4 & 0x0000FF00u) z4.y = -3.0e38f;
        if (m4 & 0x00FF0000u) z4.z = -3.0e38f;
        if (m4 & 0xFF000000u) z4.w = -3.0e38f;
        zsm4[e4] = z4;
      }
    }
  }
  __syncthreads();

  // ---------------- Phase 3: sparsemax threshold tau (Michelot, exact) ----
  // One wave handles 2 rows; 32 elements per lane held in registers.
  for (int rr = 0; rr < 2; ++rr) {
    const int row = wv * 2 + rr;
    float zr[32];
#pragma unroll
    for (int i = 0; i < 32; ++i) zr[i] = zsm[row * LK + lane + 32 * i];

    float s = 0.f; int c = 0;
#pragma unroll
    for (int i = 0; i < 32; ++i) {
      if (zr[i] > -1.0e37f) { s += zr[i]; ++c; }    // unmasked set
    }
    s = wred_sum_f(s);
    c = wred_sum_i(c);

    float tau;
    if (c == 0) {
      tau = 3.0e38f;                                // fully-masked row -> zeros
    } else {
      tau = (s - 1.0f) / (float)c;
      for (int it = 0; it < 64; ++it) {             // support strictly shrinks
        float s2 = 0.f; int c2 = 0;
#pragma unroll
        for (int i = 0; i < 32; ++i) {
          const float zz = zr[i];
          if (zz > tau) { s2 += zz; ++c2; }
        }
        s2 = wred_sum_f(s2);
        c2 = wred_sum_i(c2);
        if (c2 == c) break;                         // fixed point: tau exact
        c = c2;
        tau = (s2 - 1.0f) / (float)c2;
      }
    }
    if (lane == 0) tausm[row] = tau;
  }
  __syncthreads();

  // ------- Phase 4: P = relu(z - tau); stream attn out as NT float4 -------
  {
    v4f* attnrow4 = reinterpret_cast<v4f*>(attn + ((size_t)b * LQ + q0) * LK);
#pragma unroll 4
    for (int i = 0; i < Z_ELEMS / (4 * NTHREADS); ++i) {
      const int e4 = tid + NTHREADS * i;            // float4 index
      const int row = e4 >> 8;                      // (4*e4) >> 10
      const float tau = tausm[row];
      v4f z4 = zsm4[e4];
      z4.x = fmaxf(z4.x - tau, 0.0f);
      z4.y = fmaxf(z4.y - tau, 0.0f);
      z4.z = fmaxf(z4.z - tau, 0.0f);
      z4.w = fmaxf(z4.w - tau, 0.0f);
      zsm4[e4] = z4;
      __builtin_nontemporal_store(z4, &attnrow4[e4]);  // 268MB stream: NT
    }
  }
  __syncthreads();

  // ---------------- Phase 5: out = P @ V via WMMA f32 16x16x4 -------------
  // wave -> (n-tile = wv&3, K-half = wv>>2); 128 k-steps of 4 each.
  {
    const int nt = wv & 3;
    const int khalf = wv >> 2;
    const int d0 = nt * 16;
    const int kbase = khalf * 512;
    const float* vb = v + (size_t)b * LK * DH;
    v8f o = {0.f, 0.f, 0.f, 0.f, 0.f, 0.f, 0.f, 0.f};
    for (int s = 0; s < 128; ++s) {
      const int kk = kbase + s * 4 + 2 * kh;
      v2f aa;
      aa.x = zsm[ln * LK + kk];
      aa.y = zsm[ln * LK + kk + 1];
      v2f bb;
      bb.x = vb[(size_t)kk * DH + d0 + ln];
      bb.y = vb[(size_t)(kk + 1) * DH + d0 + ln];
      o = __builtin_amdgcn_wmma_f32_16x16x4_f32(false, aa, false, bb,
                                                (short)0, o, false, false);
    }
#pragma unroll
    for (int j = 0; j < 8; ++j) psm[wv * 256 + j * 32 + lane] = o[j];
  }
  __syncthreads();

  // Reduce the two K-halves and write out[b, q0+m, d]
  {
    float* outrow = out + ((size_t)b * LQ + q0) * DH;
#pragma unroll
    for (int i = 0; i < (QTILE * DH) / NTHREADS; ++i) {
      const int f = tid + NTHREADS * i;     // f = m*64 + d, coalesced in d
      const int m = f >> 6;
      const int d = f & 63;
      const int ntc = d >> 4;
      const int n = d & 15;
      const int j = m & 7;
      const int lidx = (m >> 3) * 16 + n;
      const float val = psm[ntc * 256 + j * 32 + lidx] +
                        psm[(ntc + 4) * 256 + j * 32 + lidx];
      outrow[m * DH + d] = val;
    }
  }
}

extern "C" void kernel_launch(void* const* d_in, const int* in_sizes, int n_in,
                              void* d_out, int out_size, void* d_ws, size_t ws_size,
                              hipStream_t stream) {
  const float* q = (const float*)d_in[0];
  const float* k = (const float*)d_in[1];
  const float* v = (const float*)d_in[2];
  const unsigned char* mask = (const unsigned char*)d_in[3];

  float* out  = (float*)d_out;                               // [64,1024,64]
  float* attn = out + (size_t)BATCH * LQ * DH;               // [64,1024,1024]

  (void)in_sizes; (void)n_in; (void)out_size; (void)d_ws; (void)ws_size;

  // Dynamic LDS is 73.8 KB (> 64 KB default limit); opt in every call
  // (idempotent, deterministic — no static guards).
  hipFuncSetAttribute(reinterpret_cast<const void*>(sparsemax_attn_kernel),
                      hipFuncAttributeMaxDynamicSharedMemorySize, SMEM_BYTES);

  dim3 grid(LQ / QTILE, BATCH);   // 64 q-tiles x 64 batches
  sparsemax_attn_kernel<<<grid, NTHREADS, SMEM_BYTES, stream>>>(q, k, v, mask,
                                                                out, attn);
}